// Model_40673340293422
// MI455X (gfx1250) — compile-verified
//
#include <hip/hip_runtime.h>

#define N_CLASS 5
#define NH 32
#define T_STEPS 2048
#define BATCH 512
#define LDS_STRIDE 34   // 32 + 2 pad: keeps float2 (8B) alignment, dodges bank conflicts

typedef __attribute__((ext_vector_type(2))) float v2f;
typedef __attribute__((ext_vector_type(8))) float v8f;

static __device__ __forceinline__ v2f make2(float a, float b) {
    v2f r; r.x = a; r.y = b; return r;
}

static __device__ __forceinline__ float fast_tanh(float x) {
#if __has_builtin(__builtin_amdgcn_tanhf)
    return __builtin_amdgcn_tanhf(x);   // v_tanh_f32 (CDNA5 TRANS op)
#else
    float e = __expf(2.0f * x);
    return 1.0f - 2.0f / (e + 1.0f);
#endif
}

// One wave (32 lanes) per block; each block owns a 16-row batch tile and runs
// the full T=2048 recurrence. WMMA f32 16x16x4, K=32 via 8 chunks.
//
// Layouts (wave32, per ISA 7.12.2):
//   A (16x4 f32):  lane m=L%16 holds rows; half=L/16 selects K pair:
//                  vgpr0=A[m][4k+2h], vgpr1=A[m][4k+2h+1]
//   B (4x16 f32):  lane n=L%16 holds cols; vgpr0=B[4k+2h][n], vgpr1=B[4k+2h+1][n]
//   C/D (16x16):   lane n=L%16 is the column; vgpr r holds row r+8*half
__global__ __launch_bounds__(32) void rnn_scan_wmma(
    const int*   __restrict__ x,      // [T, B] int32 word ids
    const float* __restrict__ emb,    // [N_WORD, 32]
    const float* __restrict__ W_rnn,  // [64, 32]: Wx rows 0..31, Wh rows 32..63
    const float* __restrict__ b_rnn,  // [32]
    const float* __restrict__ W_cls,  // [32, 5]
    const float* __restrict__ b_cls,  // [5]
    float*       __restrict__ out)    // [B, 5]
{
    __shared__ float sH[16 * LDS_STRIDE];

    const int lane  = threadIdx.x;       // 0..31
    const int m     = lane & 15;         // row (A/M) or column (B,C,D/N) index
    const int half  = lane >> 4;         // 0 or 1
    const int gbase = blockIdx.x * 16;   // batch-tile base row

    const float* Wx = W_rnn;
    const float* Wh = W_rnn + NH * NH;

    // ---- Preload loop-invariant B fragments for Wx/Wh into VGPRs ----
    v2f wx0[8], wx1[8], wh0[8], wh1[8];
#pragma unroll
    for (int kc = 0; kc < 8; ++kc) {
        const int k0 = 4 * kc + 2 * half;
        wx0[kc] = make2(Wx[k0 * NH + m],        Wx[(k0 + 1) * NH + m]);
        wx1[kc] = make2(Wx[k0 * NH + 16 + m],   Wx[(k0 + 1) * NH + 16 + m]);
        wh0[kc] = make2(Wh[k0 * NH + m],        Wh[(k0 + 1) * NH + m]);
        wh1[kc] = make2(Wh[k0 * NH + 16 + m],   Wh[(k0 + 1) * NH + 16 + m]);
    }
    const float bias0 = b_rnn[m];        // bias indexed by output column n
    const float bias1 = b_rnn[16 + m];

    // ---- h0 = 0 ----
#pragma unroll
    for (int r = 0; r < 8; ++r) {
        sH[(r + 8 * half) * LDS_STRIDE + m]      = 0.0f;
        sH[(r + 8 * half) * LDS_STRIDE + 16 + m] = 0.0f;
    }
    __syncthreads();

    // ---- Prefetch t=0 embedding A-fragments (gather from L2-resident table) ----
    int idx = x[gbase + m];
    v2f xf[8];
#pragma unroll
    for (int kc = 0; kc < 8; ++kc)
        xf[kc] = *(const v2f*)(emb + (long)idx * NH + 4 * kc + 2 * half);

    for (int t = 0; t < T_STEPS; ++t) {
        // acc = bias (broadcast along rows; column picked by lane)
        v8f acc0, acc1;
#pragma unroll
        for (int r = 0; r < 8; ++r) { acc0[r] = bias0; acc1[r] = bias1; }

        // acc += x_t @ Wx   (independent of h -> front of the step)
#pragma unroll
        for (int kc = 0; kc < 8; ++kc) {
            acc0 = __builtin_amdgcn_wmma_f32_16x16x4_f32(
                false, xf[kc], false, wx0[kc], (short)0, acc0, false, false);
            acc1 = __builtin_amdgcn_wmma_f32_16x16x4_f32(
                false, xf[kc], false, wx1[kc], (short)0, acc1, false, false);
        }

        // Prefetch next step's gather; latency hides under the h@Wh chain.
        if (t + 1 < T_STEPS) {
            idx = x[(t + 1) * BATCH + gbase + m];
#pragma unroll
            for (int kc = 0; kc < 8; ++kc)
                xf[kc] = *(const v2f*)(emb + (long)idx * NH + 4 * kc + 2 * half);
        }

        // acc += h_t @ Wh   (serial dependence: A-fragments read from LDS)
#pragma unroll
        for (int kc = 0; kc < 8; ++kc) {
            v2f ah = *(const v2f*)(&sH[m * LDS_STRIDE + 4 * kc + 2 * half]);
            acc0 = __builtin_amdgcn_wmma_f32_16x16x4_f32(
                false, ah, false, wh0[kc], (short)0, acc0, false, false);
            acc1 = __builtin_amdgcn_wmma_f32_16x16x4_f32(
                false, ah, false, wh1[kc], (short)0, acc1, false, false);
        }

        __syncthreads();   // all lanes done reading sH before overwrite

        // h_{t+1} = tanh(acc); store D layout -> LDS (rows r+8*half, col = lane)
#pragma unroll
        for (int r = 0; r < 8; ++r) {
            sH[(r + 8 * half) * LDS_STRIDE + m]      = fast_tanh(acc0[r]);
            sH[(r + 8 * half) * LDS_STRIDE + 16 + m] = fast_tanh(acc1[r]);
        }
        __syncthreads();   // writes visible before next step's reads
    }

    // ---- y = h_final @ W_cls + b_cls  (16 x 5, scalar per lane) ----
    if (lane < 16) {
#pragma unroll
        for (int c = 0; c < N_CLASS; ++c) {
            float acc = b_cls[c];
#pragma unroll
            for (int k = 0; k < NH; ++k)
                acc += sH[m * LDS_STRIDE + k] * W_cls[k * N_CLASS + c];
            out[(gbase + m) * N_CLASS + c] = acc;
        }
    }
}

extern "C" void kernel_launch(void* const* d_in, const int* in_sizes, int n_in,
                              void* d_out, int out_size, void* d_ws, size_t ws_size,
                              hipStream_t stream) {
    const int*   x     = (const int*)d_in[0];
    const float* emb   = (const float*)d_in[1];
    const float* W_rnn = (const float*)d_in[2];
    const float* b_rnn = (const float*)d_in[3];
    const float* W_cls = (const float*)d_in[4];
    const float* b_cls = (const float*)d_in[5];
    float* out = (float*)d_out;

    rnn_scan_wmma<<<dim3(BATCH / 16), dim3(32), 0, stream>>>(
        x, emb, W_rnn, b_rnn, W_cls, b_cls, out);
}